// Resampling_25426206392402
// MI455X (gfx1250) — compile-verified
//
#include <hip/hip_runtime.h>

typedef __attribute__((ext_vector_type(2))) float v2f;
typedef __attribute__((ext_vector_type(8))) float v8f;

// Problem constants from the reference: B=4, P=8, H=W=D=32, C=16
// fmap  : (32 bp, 32 h, 32 w, 32 d, 16 c) float
// theta : (32 bp, 3, 4) float
// out   : (32 bp, 32 h, 32 w, 32 d, 16 c) float
//
// Grid: 32 bp * 512 blocks = 16384 blocks of 256 threads.
// Each block handles 64 positions; each position is served by 4 lanes,
// one float4 of channels per lane (coalesced b128 gathers/stores).

__global__ __launch_bounds__(256) void Resampling_25426206392402_kernel(
    const float* __restrict__ fmap,
    const float* __restrict__ theta,
    float* __restrict__ out)
{
    __shared__ float sc[64 * 3];   // (y,x,z) per local position

    const int bp      = blockIdx.x >> 9;          // 512 blocks per (b,p)
    const int posbase = (blockIdx.x & 511) << 6;  // 64 positions per block
    const int tid     = threadIdx.x;

    // ---- Prologue: wave 0 computes coords for the block's 64 positions ----
    // einsum('ij,jn->in', theta[bp] (3x4), grid (4x16)) on the tensor core:
    //   A (16x4)  = theta zero-padded             (lane M, VGPR0/1 = K0/K1 or K2/K3)
    //   B (4x16)  = [h; d; w; 1] for 16 positions (VGPR0 = rows K0|K2, VGPR1 = K1|K3)
    //   D (16x16) rows 0..2 = y_s, x_s, z_s for the 16 positions (N = lane 0..15)
    if (tid < 32) {
        const int  lane = tid;
        const int  m    = lane & 15;
        const bool lo   = lane < 16;

        float a0 = 0.0f, a1 = 0.0f;
        if (m < 3) {
            const float* th = theta + bp * 12 + m * 4 + (lo ? 0 : 2);
            a0 = th[0];
            a1 = th[1];
        }
        v2f A;
        A[0] = a0;
        A[1] = a1;

        #pragma unroll
        for (int it = 0; it < 4; ++it) {
            const int   pl = posbase + it * 16 + m;     // this lane's column position
            const float hh = (float)((pl >> 10) & 31);
            const float ww = (float)((pl >> 5) & 31);
            const float dd = (float)(pl & 31);

            v2f Bm;
            Bm[0] = lo ? hh : ww;    // K=0 row: h   | K=2 row: w
            Bm[1] = lo ? dd : 1.0f;  // K=1 row: d   | K=3 row: 1
            v8f Cm = {};

            v8f Dm = __builtin_amdgcn_wmma_f32_16x16x4_f32(
                false, A, false, Bm, (short)0, Cm, false, false);

            if (lo) {
                const int li = it * 16 + m;
                sc[li * 3 + 0] = Dm[0];  // y_s
                sc[li * 3 + 1] = Dm[1];  // x_s
                sc[li * 3 + 2] = Dm[2];  // z_s
            }
        }
    }
    __syncthreads();

    // ---- Main phase: trilinear gather, 4 lanes per position ----
    const int q    = tid & 3;        // channel quad (4 floats)
    const int pidx = tid >> 2;       // 0..63 local position
    const int pos  = posbase + pidx; // position within (b,p)

    // padded coordinates (reference pads by 2 and adds 2)
    const float y = sc[pidx * 3 + 0] + 2.0f;
    const float x = sc[pidx * 3 + 1] + 2.0f;
    const float z = sc[pidx * 3 + 2] + 2.0f;

    int y0 = (int)floorf(y); y0 = min(max(y0, 0), 34);
    int x0 = (int)floorf(x); x0 = min(max(x0, 0), 34);
    int z0 = (int)floorf(z); z0 = min(max(z0, 0), 34);

    const float yd = y - (float)y0, xd = x - (float)x0, zd = z - (float)z0;
    const float yd1 = 1.0f - yd, xd1 = 1.0f - xd, zd1 = 1.0f - zd;

    const float* src = fmap + (size_t)bp * (32 * 32 * 32 * 16);

    // Branchless gather: clamp corner indices into the valid volume so every
    // load is in-bounds, and zero the corner's weight when it falls in the
    // zero-pad region (identical math: pad value is 0).
    float  wgt[8];
    int    idx[8];
    #pragma unroll
    for (int k = 0; k < 8; ++k) {
        const int dy = (k >> 2) & 1, dx = (k >> 1) & 1, dz = k & 1;
        const int oy = y0 + dy - 2, ox = x0 + dx - 2, oz = z0 + dz - 2;
        const bool valid = ((unsigned)oy < 32u) &&
                           ((unsigned)ox < 32u) &&
                           ((unsigned)oz < 32u);
        const float w = (dy ? yd : yd1) * (dx ? xd : xd1) * (dz ? zd : zd1);
        wgt[k] = valid ? w : 0.0f;
        const int cy = min(max(oy, 0), 31);
        const int cx = min(max(ox, 0), 31);
        const int cz = min(max(oz, 0), 31);
        idx[k] = ((cy << 10) + (cx << 5) + cz) * 16 + (q << 2);
    }

    float4 c[8];
    #pragma unroll
    for (int k = 0; k < 8; ++k)
        c[k] = *(const float4*)(src + idx[k]);   // 8 unconditional b128 gathers

    float4 acc = make_float4(0.0f, 0.0f, 0.0f, 0.0f);
    #pragma unroll
    for (int k = 0; k < 8; ++k) {
        acc.x = fmaf(wgt[k], c[k].x, acc.x);
        acc.y = fmaf(wgt[k], c[k].y, acc.y);
        acc.z = fmaf(wgt[k], c[k].z, acc.z);
        acc.w = fmaf(wgt[k], c[k].w, acc.w);
    }

    float4* op = (float4*)(out + ((size_t)bp * 32768 + (size_t)pos) * 16 + (q << 2));
    *op = acc;
}

extern "C" void kernel_launch(void* const* d_in, const int* in_sizes, int n_in,
                              void* d_out, int out_size, void* d_ws, size_t ws_size,
                              hipStream_t stream) {
    const float* fmap  = (const float*)d_in[0];  // (4,8,32,32,32,16) f32
    const float* theta = (const float*)d_in[1];  // (4,8,3,4) f32
    float*       out   = (float*)d_out;          // (4,8,32,32,32,16) f32

    // 32 (b,p) pairs * 512 blocks each; 256 threads = 64 positions per block
    dim3 grid(32 * 512);
    dim3 block(256);
    Resampling_25426206392402_kernel<<<grid, block, 0, stream>>>(fmap, theta, out);
}